// UCSAGE_32375463477418
// MI455X (gfx1250) — compile-verified
//
#include <hip/hip_runtime.h>
#include <cstdint>
#include <cstddef>

typedef __attribute__((ext_vector_type(2))) float v2f;
typedef __attribute__((ext_vector_type(8))) float v8f;

// ---------------------------------------------------------------------------
// Zero-fill
// ---------------------------------------------------------------------------
__global__ void ucsage_zero_f32(float* __restrict__ p, long long n) {
    long long i = (long long)blockIdx.x * blockDim.x + threadIdx.x;
    if (i < n) p[i] = 0.0f;
}

// ---------------------------------------------------------------------------
// Degree count: cnt[dst] += 1  (native f32 L2 atomics)
// ---------------------------------------------------------------------------
__global__ void ucsage_degree(const long long* __restrict__ dst,
                              float* __restrict__ cnt, int E) {
    int e = blockIdx.x * blockDim.x + threadIdx.x;
    if (e < E) {
        int d = (int)dst[e];
        unsafeAtomicAdd(cnt + d, 1.0f);
    }
}

// ---------------------------------------------------------------------------
// Edge scatter-add: agg[dst][:] += x[src][:]
// F/4 lanes per edge, float4 gather, 4x global_atomic_add_f32 scatter.
// Gathers + atomics stay resident in the 192MB L2 (feature matrix <= 12.8MB).
// ---------------------------------------------------------------------------
template <int F>
__global__ void ucsage_scatter(const float* __restrict__ x,
                               const long long* __restrict__ src,
                               const long long* __restrict__ dst,
                               float* __restrict__ agg, int E) {
    constexpr int Q = F / 4;
    long long tid = (long long)blockIdx.x * blockDim.x + threadIdx.x;
    int e = (int)(tid / Q);
    int q = (int)(tid - (long long)e * Q);
    if (e >= E) return;
    int s = (int)src[e];
    int d = (int)dst[e];
    const float4 v = *(const float4*)(x + (size_t)s * F + q * 4);
    float* p = agg + (size_t)d * F + q * 4;
    unsafeAtomicAdd(p + 0, v.x);
    unsafeAtomicAdd(p + 1, v.y);
    unsafeAtomicAdd(p + 2, v.z);
    unsafeAtomicAdd(p + 3, v.w);
}

// ---------------------------------------------------------------------------
// Node update via full-fp32 WMMA:
//   out = act( [agg/cnt , x] @ [Wl ; Wr] + b )
// One wave per 16-node tile. K = 2*F_IN reduced 4 at a time with
// V_WMMA_F32_16X16X4_F32. Weights + bias staged in LDS.
//
// A 16x4 f32 fragment (per ISA): lanes 0-15 hold M=lane, {K=k0,k0+1};
// lanes 16-31 hold M=lane-16, {K=k0+2,k0+3}.
// D 16x16 f32: VGPR r -> M = r + 8*(lane>=16), N = lane&15.
// ---------------------------------------------------------------------------
template <int F_IN, int F_OUT, int ACT> // ACT: 0 = relu, 1 = sigmoid
__global__ __launch_bounds__(256) void ucsage_node_wmma(
    const float* __restrict__ x, const float* __restrict__ agg,
    const float* __restrict__ cnt,
    const float* __restrict__ Wl, const float* __restrict__ Wr,
    const float* __restrict__ bias, float* __restrict__ out, int N) {
    constexpr int K = 2 * F_IN;
    constexpr int NT = (F_OUT + 15) / 16; // column tiles of 16

    __shared__ float sW[K * F_OUT]; // combined [Wl ; Wr], k-major
    __shared__ float sB[F_OUT];
    for (int i = threadIdx.x; i < K * F_OUT; i += blockDim.x)
        sW[i] = (i < F_IN * F_OUT) ? Wl[i] : Wr[i - F_IN * F_OUT];
    for (int i = threadIdx.x; i < F_OUT; i += blockDim.x) sB[i] = bias[i];
    __syncthreads();

    const int lane = threadIdx.x & 31;
    const int wave = threadIdx.x >> 5;
    const int tile = blockIdx.x * 8 + wave;
    const int nTiles = (N + 15) >> 4;
    if (tile >= nTiles) return; // wave-uniform: EXEC stays all-ones for WMMA

    const int half = lane >> 4;  // 0: K=k0..k0+1, 1: K=k0+2..k0+3
    const int l16  = lane & 15;

    // A-side row for this lane
    int rowNode = tile * 16 + l16;
    if (rowNode >= N) rowNode = N - 1; // clamp (loads only; stores guarded)
    const float ic = 1.0f / fmaxf(cnt[rowNode], 1.0f);
    const float* aggRow = agg + (size_t)rowNode * F_IN;
    const float* xRow   = x   + (size_t)rowNode * F_IN;

    v8f acc[NT];
#pragma unroll
    for (int t = 0; t < NT; ++t) acc[t] = v8f{0, 0, 0, 0, 0, 0, 0, 0};

#pragma unroll
    for (int k0 = 0; k0 < K; k0 += 4) {
        const int ka = k0 + half * 2; // even; ka,ka+1 on same side of F_IN
        v2f a;
        if (ka < F_IN) {
            a[0] = aggRow[ka] * ic;
            a[1] = aggRow[ka + 1] * ic;
        } else {
            a[0] = xRow[ka - F_IN];
            a[1] = xRow[ka - F_IN + 1];
        }
#pragma unroll
        for (int t = 0; t < NT; ++t) {
            const int c  = t * 16 + l16;
            const int cc = (c < F_OUT) ? c : 0;
            const float m = (c < F_OUT) ? 1.0f : 0.0f;
            v2f b;
            b[0] = sW[ka * F_OUT + cc] * m;
            b[1] = sW[(ka + 1) * F_OUT + cc] * m;
            acc[t] = __builtin_amdgcn_wmma_f32_16x16x4_f32(
                false, a, false, b, (short)0, acc[t], false, false);
        }
    }

    // Bias + activation + store
#pragma unroll
    for (int t = 0; t < NT; ++t) {
        const int c = t * 16 + l16;
        if (c >= F_OUT) continue;
        const float bv = sB[c];
#pragma unroll
        for (int r = 0; r < 8; ++r) {
            const int mrow = half * 8 + r;
            const int node = tile * 16 + mrow;
            if (node < N) {
                float v = acc[t][r] + bv;
                v = (ACT == 0) ? fmaxf(v, 0.0f) : (1.0f / (1.0f + expf(-v)));
                out[(size_t)node * F_OUT + c] = v;
            }
        }
    }
}

// ---------------------------------------------------------------------------
// Launch
// ---------------------------------------------------------------------------
static inline int cdiv_ll(long long a, long long b) { return (int)((a + b - 1) / b); }

extern "C" void kernel_launch(void* const* d_in, const int* in_sizes, int n_in,
                              void* d_out, int out_size, void* d_ws, size_t ws_size,
                              hipStream_t stream) {
    const float* x = (const float*)d_in[0];
    const long long* ei = (const long long*)d_in[1];
    const float* Wl1 = (const float*)d_in[2];
    const float* Wr1 = (const float*)d_in[3];
    const float* b1  = (const float*)d_in[4];
    const float* Wl2 = (const float*)d_in[5];
    const float* Wr2 = (const float*)d_in[6];
    const float* b2  = (const float*)d_in[7];
    const float* Wl3 = (const float*)d_in[8];
    const float* Wr3 = (const float*)d_in[9];
    const float* b3  = (const float*)d_in[10];

    constexpr int IN = 32, HID = 24, OUT = 12;
    const int N = in_sizes[0] / IN;
    const int E = in_sizes[1] / 2;
    const long long* src = ei;
    const long long* dst = ei + E;

    // Workspace layout
    char* ws = (char*)d_ws;
    size_t off = 0;
    float* cnt = (float*)(ws + off); off += (size_t)N * 4;        off = (off + 255) & ~(size_t)255;
    float* agg = (float*)(ws + off); off += (size_t)N * IN * 4;   off = (off + 255) & ~(size_t)255;
    float* h1  = (float*)(ws + off); off += (size_t)N * HID * 4;  off = (off + 255) & ~(size_t)255;
    float* h2  = (float*)(ws + off);
    float* out = (float*)d_out;

    const int tiles = (N + 15) / 16;
    const int nodeBlocks = (tiles + 7) / 8; // 8 waves (16-node tiles) per block

    // Degree (shared by all layers)
    ucsage_zero_f32<<<cdiv_ll(N, 256), 256, 0, stream>>>(cnt, N);
    ucsage_degree<<<cdiv_ll(E, 256), 256, 0, stream>>>(dst, cnt, E);

    // Layer 1: 32 -> 24, relu
    ucsage_zero_f32<<<cdiv_ll((long long)N * IN, 256), 256, 0, stream>>>(agg, (long long)N * IN);
    ucsage_scatter<IN><<<cdiv_ll((long long)E * (IN / 4), 256), 256, 0, stream>>>(x, src, dst, agg, E);
    ucsage_node_wmma<IN, HID, 0><<<nodeBlocks, 256, 0, stream>>>(x, agg, cnt, Wl1, Wr1, b1, h1, N);

    // Layer 2: 24 -> 24, relu
    ucsage_zero_f32<<<cdiv_ll((long long)N * HID, 256), 256, 0, stream>>>(agg, (long long)N * HID);
    ucsage_scatter<HID><<<cdiv_ll((long long)E * (HID / 4), 256), 256, 0, stream>>>(h1, src, dst, agg, E);
    ucsage_node_wmma<HID, HID, 0><<<nodeBlocks, 256, 0, stream>>>(h1, agg, cnt, Wl2, Wr2, b2, h2, N);

    // Layer 3: 24 -> 12, sigmoid
    ucsage_zero_f32<<<cdiv_ll((long long)N * HID, 256), 256, 0, stream>>>(agg, (long long)N * HID);
    ucsage_scatter<HID><<<cdiv_ll((long long)E * (HID / 4), 256), 256, 0, stream>>>(h2, src, dst, agg, E);
    ucsage_node_wmma<HID, OUT, 1><<<nodeBlocks, 256, 0, stream>>>(h2, agg, cnt, Wl3, Wr3, b3, out, N);
}